// GN_81784767250539
// MI455X (gfx1250) — compile-verified
//
#include <hip/hip_runtime.h>
#include <hip/hip_bf16.h>

typedef __attribute__((ext_vector_type(2))) float v2f;
typedef __attribute__((ext_vector_type(8))) float v8f;

#define DFEAT 96
#define PI_F 3.14159265358979323846f

// ---------------------------------------------------------------------------
// Kernel 1: zero the h_e accumulator workspace (float4-vectorized).
// ---------------------------------------------------------------------------
__global__ void GN_zero_kernel(float4* ws, int n4) {
    int i = blockIdx.x * blockDim.x + threadIdx.x;
    if (i < n4) ws[i] = make_float4(0.f, 0.f, 0.f, 0.f);
}

// ---------------------------------------------------------------------------
// Kernel 2: edge scatter-add: he[dst] += h[src], 4 features per thread.
// 24 threads cover one edge's 96 features; lanes are coalesced over features.
// ---------------------------------------------------------------------------
__global__ void GN_scatter_kernel(const float* __restrict__ h,
                                  const int* __restrict__ src,
                                  const int* __restrict__ dst,
                                  float* __restrict__ he,
                                  long long total /* n_edges * 24 */) {
    long long idx = (long long)blockIdx.x * blockDim.x + threadIdx.x;
    if (idx >= total) return;
    int e  = (int)(idx / 24);
    int d4 = (int)(idx % 24) * 4;
    int s = src[e];
    int t = dst[e];
    float4 v = *(const float4*)(h + (long long)s * DFEAT + d4);
    float* out = he + (long long)t * DFEAT + d4;
    unsafeAtomicAdd(out + 0, v.x);
    unsafeAtomicAdd(out + 1, v.y);
    unsafeAtomicAdd(out + 2, v.z);
    unsafeAtomicAdd(out + 3, v.w);
}

// ---------------------------------------------------------------------------
// Kernel 3: out = (h + pi*he) @ W^T + b via V_WMMA_F32_16X16X4_F32.
// One wave computes one 16x16 output tile; K loop = 24 wmma steps of K=4.
//
// A (16x4 f32) layout: lane L holds row M=L%16; VGPR0/1 = K = 2*(L/16)+{0,1}.
// B (4x16 f32) layout: lane L holds col N=L%16; VGPR0/1 = K = 2*(L/16)+{0,1}.
//   B[k][n] = W[n][k]  (torch Linear: y = x @ W^T), so B loads are contiguous
//   float2 reads from row n of W.
// C/D (16x16 f32, 8 VGPRs): reg r, lane L -> row = r + 8*(L/16), col = L%16.
// ---------------------------------------------------------------------------
__global__ void GN_gemm_wmma_kernel(const float* __restrict__ h,
                                    const float* __restrict__ he,
                                    const float* __restrict__ W,
                                    const float* __restrict__ b,
                                    float* __restrict__ out,
                                    int n_tiles /* mtiles * 6 */) {
    const int wave = (blockIdx.x * blockDim.x + threadIdx.x) >> 5;
    if (wave >= n_tiles) return;                 // wave-uniform: EXEC stays all-1s

    const int lane = threadIdx.x & 31;
    const int mtile = wave / 6;
    const int ntile = wave % 6;
    const int m0 = mtile * 16;
    const int n0 = ntile * 16;

    const int half = lane >> 4;                  // selects K pair 0/1 vs 2/3
    const int lm   = lane & 15;                  // M index for A, N index for B

    // Accumulator initialized with bias (bias depends only on column N).
    const float bv = b[n0 + lm];
    v8f c;
#pragma unroll
    for (int r = 0; r < 8; ++r) c[r] = bv;

    const v2f* hrow = (const v2f*)(h  + (long long)(m0 + lm) * DFEAT);
    const v2f* erow = (const v2f*)(he + (long long)(m0 + lm) * DFEAT);
    const v2f* wrow = (const v2f*)(W  + (long long)(n0 + lm) * DFEAT);

#pragma unroll
    for (int k0 = 0; k0 < DFEAT; k0 += 4) {
        const int kv = (k0 >> 1) + half;         // float2 index: k = k0 + 2*half
        v2f hv = hrow[kv];
        v2f ev = erow[kv];
        v2f a;
        a.x = __builtin_fmaf(PI_F, ev.x, hv.x);  // x = h + pi * h_e, fused
        a.y = __builtin_fmaf(PI_F, ev.y, hv.y);
        v2f bb = wrow[kv];
        // (neg_a, A, neg_b, B, c_mod, C, reuse_a, reuse_b)
        c = __builtin_amdgcn_wmma_f32_16x16x4_f32(false, a, false, bb,
                                                  (short)0, c, false, false);
    }

    // Store the 16x16 tile (N = 96 exactly divides into 6 tiles; M = 50000 is
    // exactly 3125 tiles, so no edge masking needed).
    float* obase = out + (long long)(m0 + 8 * half) * DFEAT + n0 + lm;
#pragma unroll
    for (int r = 0; r < 8; ++r) {
        obase[(long long)r * DFEAT] = c[r];
    }
}

// ---------------------------------------------------------------------------
// Launch: inputs are (h, src, dst, W, b) in setup_inputs() order.
// d_ws holds h_e: n_nodes * 96 floats = 19.2 MB.
// ---------------------------------------------------------------------------
extern "C" void kernel_launch(void* const* d_in, const int* in_sizes, int n_in,
                              void* d_out, int out_size, void* d_ws, size_t ws_size,
                              hipStream_t stream) {
    const float* h   = (const float*)d_in[0];
    const int*   src = (const int*)d_in[1];
    const int*   dst = (const int*)d_in[2];
    const float* W   = (const float*)d_in[3];
    const float* b   = (const float*)d_in[4];
    float* out = (float*)d_out;
    float* he  = (float*)d_ws;

    const int n_nodes = in_sizes[0] / DFEAT;   // 50000
    const int n_edges = in_sizes[1];           // 800000

    // 1) zero h_e accumulator
    {
        int n4 = (n_nodes * DFEAT) / 4;
        int blocks = (n4 + 255) / 256;
        GN_zero_kernel<<<blocks, 256, 0, stream>>>((float4*)he, n4);
    }
    // 2) scatter-add over edges
    {
        long long total = (long long)n_edges * 24;
        int blocks = (int)((total + 255) / 256);
        GN_scatter_kernel<<<blocks, 256, 0, stream>>>(h, src, dst, he, total);
    }
    // 3) fused (h + pi*he) @ W^T + b with fp32 WMMA
    {
        const int mtiles = n_nodes / 16;       // 3125 exactly
        const int n_tiles = mtiles * 6;        // 18750 tiles (16x16 each)
        const int waves_per_block = 8;         // 256 threads, wave32
        int blocks = (n_tiles + waves_per_block - 1) / waves_per_block;
        GN_gemm_wmma_kernel<<<blocks, 256, 0, stream>>>(h, he, W, b, out, n_tiles);
    }
}